// AutoregressiveDecoder_16209206575633
// MI455X (gfx1250) — compile-verified
//
#include <hip/hip_runtime.h>
#include <math.h>

#define Bn   256
#define Nn   1000
#define Dn   256
#define Hn   8
#define DHn  32
#define NEGV (-1000000000.0f)

typedef float v2f __attribute__((ext_vector_type(2)));
typedef float v8f __attribute__((ext_vector_type(8)));

// ---------------------------------------------------------------------------
// K0: X[b, 0:256] = node_embeddings[b, head[b], :]; X[b, 256:512] = graph_context[b,:]
// ---------------------------------------------------------------------------
__global__ void k_gather(const float* __restrict__ node_emb,
                         const float* __restrict__ gctx,
                         const int*   __restrict__ head,
                         float*       __restrict__ X) {
    int b = blockIdx.x, t = threadIdx.x;           // block 256
    int hsel = head[b];
    X[b * 512 + t]       = node_emb[(size_t)b * Nn * Dn + (size_t)hsel * Dn + t];
    X[b * 512 + 256 + t] = gctx[b * Dn + t];
}

// ---------------------------------------------------------------------------
// K1: step_context (256x512) = X (256x512) @ W_ctx[0:512,:] + dyn @ W_ctx[512:515,:]
// WMMA f32 16x16x4. One wave per 16x16 tile. grid (16 Mtiles, 8), block 128 (4 waves).
// A 16x4 layout: lanes 0-15 hold K=(k0,k0+1), lanes 16-31 hold K=(k0+2,k0+3), M=lane&15.
// B 4x16 layout: mirrored (lane half selects K pair), N=lane&15.
// C/D: VGPR i -> M = i + (lane<16?0:8), N = lane&15.
// ---------------------------------------------------------------------------
__global__ void k_ctx_gemm(const float* __restrict__ X,
                           const float* __restrict__ Wc,   // (515,512) row-major
                           const float* __restrict__ cap,
                           const float* __restrict__ dem,
                           const float* __restrict__ rem,
                           float*       __restrict__ SC) {
    int wave = threadIdx.x >> 5, lane = threadIdx.x & 31;
    int mt = blockIdx.x;                 // 0..15
    int nt = blockIdx.y * 4 + wave;      // 0..31
    int r0 = mt * 16, c0 = nt * 16;
    int lane15 = lane & 15;
    int kOff   = (lane < 16) ? 0 : 2;
    int col    = c0 + lane15;
    const float* arow = X + (size_t)(r0 + lane15) * 512 + kOff;

    v8f c = {};
    for (int k0 = 0; k0 < 512; k0 += 4) {
        v2f a = *(const v2f*)(arow + k0);                       // b64 load
        v2f bm;
        bm.x = Wc[(size_t)(k0 + kOff) * 512 + col];
        bm.y = Wc[(size_t)(k0 + kOff + 1) * 512 + col];
        c = __builtin_amdgcn_wmma_f32_16x16x4_f32(false, a, false, bm,
                                                  (short)0, c, false, false);
    }
    // dyn contribution: rows 512..514 of W_ctx times (capacity, demand, nodes)
    float w0 = Wc[(size_t)512 * 512 + col];
    float w1 = Wc[(size_t)513 * 512 + col];
    float w2 = Wc[(size_t)514 * 512 + col];
    int mbase = r0 + ((lane < 16) ? 0 : 8);
#pragma unroll
    for (int i = 0; i < 8; ++i) {
        int row = mbase + i;
        SC[(size_t)row * 512 + col] =
            c[i] + cap[row] * w0 + dem[row] * w1 + rem[row] * w2;
    }
}

// ---------------------------------------------------------------------------
// K2: dual-query fused MHA. One block per batch, 512 threads = 16 waves.
// wave -> head h = wave/2, partition p = wave&1. Within a wave: 4 nodes/iter,
// 8 lanes per node (lane = g*8+s). K row read once as float4, used for BOTH
// queries; V read lane-per-dim (coalesced); online softmax per query.
// ---------------------------------------------------------------------------
__global__ __launch_bounds__(512) void k_mha(const float* __restrict__ SC,
                                             const float* __restrict__ K,
                                             const float* __restrict__ V,
                                             const int*   __restrict__ feas,
                                             float*       __restrict__ mha1,
                                             float*       __restrict__ mha2) {
    __shared__ __align__(16) float q1s[256];
    __shared__ __align__(16) float q2s[256];
    __shared__ float sm[2][16], sl[2][16];
    __shared__ __align__(16) float sacc[2][16][32];
    __shared__ int anyFeas;

    int b = blockIdx.x, tid = threadIdx.x;
    if (tid == 0) anyFeas = 0;
    if (tid < 256) q1s[tid] = SC[b * 512 + tid];
    else           q2s[tid - 256] = SC[b * 512 + tid];
    __syncthreads();
    bool found = false;
    for (int n = tid; n < Nn; n += 512) found |= (feas[b * Nn + n] != 0);
    if (found) anyFeas = 1;
    __syncthreads();
    bool noFeas = (anyFeas == 0);

    int wave = tid >> 5, lane = tid & 31;
    int h = wave >> 1, p = wave & 1;
    int g = lane >> 3, s = lane & 7;

    const float4 q1r = *(const float4*)&q1s[h * DHn + s * 4];
    const float4 q2r = *(const float4*)&q2s[h * DHn + s * 4];
    const float* Kb = K + (size_t)(b * Hn + h) * Nn * DHn;
    const float* Vb = V + (size_t)(b * Hn + h) * Nn * DHn;
    const int*   fb = feas + b * Nn;

    const float scale = 0.17677669529663687f;   // 1/sqrt(32)
    float m1 = -INFINITY, l1 = 0.f, a1 = 0.f;
    float m2 = -INFINITY, l2 = 0.f, a2 = 0.f;

    for (int base = p * 4; base < Nn; base += 8) {     // 125 iters, exact
        int n = base + g;
        bool ok = (fb[n] != 0) || (n == 0 && noFeas);
        const float4 kv = *(const float4*)(Kb + (size_t)n * DHn + s * 4);
        float d1 = kv.x * q1r.x + kv.y * q1r.y + kv.z * q1r.z + kv.w * q1r.w;
        float d2 = kv.x * q2r.x + kv.y * q2r.y + kv.z * q2r.z + kv.w * q2r.w;
        d1 += __shfl_xor(d1, 1, 32); d1 += __shfl_xor(d1, 2, 32); d1 += __shfl_xor(d1, 4, 32);
        d2 += __shfl_xor(d2, 1, 32); d2 += __shfl_xor(d2, 2, 32); d2 += __shfl_xor(d2, 4, 32);
        float s1 = ok ? d1 * scale : NEGV;
        float s2 = ok ? d2 * scale : NEGV;

        float sc1[4], sc2[4], vj[4];
#pragma unroll
        for (int j = 0; j < 4; ++j) {
            sc1[j] = __shfl(s1, j * 8, 32);
            sc2[j] = __shfl(s2, j * 8, 32);
            vj[j]  = Vb[(size_t)(base + j) * DHn + lane];   // coalesced 128B
        }
        // online softmax, query 1
        float mx = fmaxf(fmaxf(sc1[0], sc1[1]), fmaxf(sc1[2], sc1[3]));
        float mn = fmaxf(m1, mx);
        float r  = __expf(m1 - mn);
        float e0 = __expf(sc1[0] - mn), e1 = __expf(sc1[1] - mn);
        float e2 = __expf(sc1[2] - mn), e3 = __expf(sc1[3] - mn);
        l1 = l1 * r + e0 + e1 + e2 + e3;
        a1 = a1 * r + e0 * vj[0] + e1 * vj[1] + e2 * vj[2] + e3 * vj[3];
        m1 = mn;
        // online softmax, query 2
        mx = fmaxf(fmaxf(sc2[0], sc2[1]), fmaxf(sc2[2], sc2[3]));
        mn = fmaxf(m2, mx);
        r  = __expf(m2 - mn);
        e0 = __expf(sc2[0] - mn); e1 = __expf(sc2[1] - mn);
        e2 = __expf(sc2[2] - mn); e3 = __expf(sc2[3] - mn);
        l2 = l2 * r + e0 + e1 + e2 + e3;
        a2 = a2 * r + e0 * vj[0] + e1 * vj[1] + e2 * vj[2] + e3 * vj[3];
        m2 = mn;
    }

    if (lane == 0) { sm[0][wave] = m1; sl[0][wave] = l1; sm[1][wave] = m2; sl[1][wave] = l2; }
    sacc[0][wave][lane] = a1;
    sacc[1][wave][lane] = a2;
    __syncthreads();

    if (p == 0) {                       // merge the two partitions of this head
#pragma unroll
        for (int q = 0; q < 2; ++q) {
            float ma = sm[q][wave], mb = sm[q][wave + 1];
            float mf = fmaxf(ma, mb);
            float ea = __expf(ma - mf), eb = __expf(mb - mf);
            float lf = sl[q][wave] * ea + sl[q][wave + 1] * eb;
            float af = sacc[q][wave][lane] * ea + sacc[q][wave + 1][lane] * eb;
            float* dst = (q == 0) ? mha1 : mha2;
            dst[(size_t)b * Dn + h * DHn + lane] = af / lf;
        }
    }
}

// ---------------------------------------------------------------------------
// K3: two 256x256x256 GEMMs (glimpse = mha1 @ W_out ; head_enc = mha2 @ W_sa),
// selected by blockIdx.z. WMMA f32 16x16x4.
// ---------------------------------------------------------------------------
__global__ void k_gemm256(const float* __restrict__ A0, const float* __restrict__ W0,
                          float* __restrict__ O0,
                          const float* __restrict__ A1, const float* __restrict__ W1,
                          float* __restrict__ O1) {
    const float* A = blockIdx.z ? A1 : A0;
    const float* W = blockIdx.z ? W1 : W0;
    float*       O = blockIdx.z ? O1 : O0;
    int wave = threadIdx.x >> 5, lane = threadIdx.x & 31;
    int mt = blockIdx.x;                 // 0..15
    int nt = blockIdx.y * 4 + wave;      // 0..15
    int r0 = mt * 16, c0 = nt * 16;
    int lane15 = lane & 15;
    int kOff   = (lane < 16) ? 0 : 2;
    int col    = c0 + lane15;
    const float* arow = A + (size_t)(r0 + lane15) * 256 + kOff;

    v8f c = {};
    for (int k0 = 0; k0 < 256; k0 += 4) {
        v2f a = *(const v2f*)(arow + k0);
        v2f bm;
        bm.x = W[(size_t)(k0 + kOff) * 256 + col];
        bm.y = W[(size_t)(k0 + kOff + 1) * 256 + col];
        c = __builtin_amdgcn_wmma_f32_16x16x4_f32(false, a, false, bm,
                                                  (short)0, c, false, false);
    }
    int mbase = r0 + ((lane < 16) ? 0 : 8);
#pragma unroll
    for (int i = 0; i < 8; ++i)
        O[(size_t)(mbase + i) * 256 + col] = c[i];
}

// ---------------------------------------------------------------------------
// K4: logits[b,n] = 10*tanh((glimpse[b]·logit_key[b,n])/16), mask, log_softmax.
// One block per batch, 512 threads; 8 lanes per node (coalesced float4 reads).
// ---------------------------------------------------------------------------
__global__ __launch_bounds__(512) void k_logits(const float* __restrict__ glimpse,
                                                const float* __restrict__ LK,
                                                const int*   __restrict__ feas,
                                                float*       __restrict__ logp) {
    __shared__ __align__(16) float gs[256];
    __shared__ float lg[Nn];
    __shared__ float red[512];
    __shared__ int anyFeas;

    int b = blockIdx.x, tid = threadIdx.x;
    if (tid == 0) anyFeas = 0;
    if (tid < 256) gs[tid] = glimpse[b * Dn + tid];
    __syncthreads();
    bool found = false;
    for (int n = tid; n < Nn; n += 512) found |= (feas[b * Nn + n] != 0);
    if (found) anyFeas = 1;
    __syncthreads();
    bool noFeas = (anyFeas == 0);

    int wave = tid >> 5, lane = tid & 31;
    int g = lane >> 3, s = lane & 7;
    const float* LKb = LK + (size_t)b * Nn * Dn;
    const int*   fb  = feas + b * Nn;

    float4 gr[8];
#pragma unroll
    for (int i = 0; i < 8; ++i) gr[i] = *(const float4*)&gs[i * 32 + s * 4];

    for (int base = wave * 4; base < Nn; base += 64) {
        int n = base + g;                                   // provably < 1000
        const float* row = LKb + (size_t)n * Dn;
        float part = 0.f;
#pragma unroll
        for (int i = 0; i < 8; ++i) {
            float4 lk = *(const float4*)(row + i * 32 + s * 4);
            part += lk.x * gr[i].x + lk.y * gr[i].y + lk.z * gr[i].z + lk.w * gr[i].w;
        }
        part += __shfl_xor(part, 1, 32);
        part += __shfl_xor(part, 2, 32);
        part += __shfl_xor(part, 4, 32);
        if (s == 0) {
            bool ok = (fb[n] != 0) || (n == 0 && noFeas);
            float lgt = 10.0f * tanhf(part * 0.0625f);      // /sqrt(256)
            lg[n] = ok ? lgt : NEGV;
        }
    }
    __syncthreads();

    float mx = -INFINITY;
    for (int n = tid; n < Nn; n += 512) mx = fmaxf(mx, lg[n]);
    red[tid] = mx; __syncthreads();
    for (int st = 256; st > 0; st >>= 1) {
        if (tid < st) red[tid] = fmaxf(red[tid], red[tid + st]);
        __syncthreads();
    }
    float m = red[0]; __syncthreads();

    float sum = 0.f;
    for (int n = tid; n < Nn; n += 512) sum += __expf(lg[n] - m);
    red[tid] = sum; __syncthreads();
    for (int st = 256; st > 0; st >>= 1) {
        if (tid < st) red[tid] += red[tid + st];
        __syncthreads();
    }
    float lse = m + __logf(red[0]);

    for (int n = tid; n < Nn; n += 512)
        logp[(size_t)b * Nn + n] = lg[n] - lse;
}

// ---------------------------------------------------------------------------
extern "C" void kernel_launch(void* const* d_in, const int* in_sizes, int n_in,
                              void* d_out, int out_size, void* d_ws, size_t ws_size,
                              hipStream_t stream) {
    const float* node_emb = (const float*)d_in[0];
    const float* gctx     = (const float*)d_in[1];
    const float* gkey     = (const float*)d_in[2];
    const float* gval     = (const float*)d_in[3];
    const float* lkey     = (const float*)d_in[4];
    const float* cap      = (const float*)d_in[5];
    const float* dem      = (const float*)d_in[6];
    const float* rem      = (const float*)d_in[7];
    const float* Wc       = (const float*)d_in[8];
    const float* Wo       = (const float*)d_in[9];
    const float* Wsa      = (const float*)d_in[10];
    const int*   head     = (const int*)d_in[11];
    const int*   feas     = (const int*)d_in[12];

    float* out  = (float*)d_out;
    float* logp = out;                  // B*N
    float* henc = out + (size_t)Bn * Nn; // B*D

    float* ws      = (float*)d_ws;
    float* X       = ws;                // 256*512 floats (reused as glimpse)
    float* SC      = ws + 131072;       // 256*512
    float* m1      = ws + 262144;       // 256*256
    float* m2      = ws + 327680;       // 256*256
    float* glimpse = X;                 // reuse after k_ctx_gemm consumes X

    k_gather  <<<Bn, 256, 0, stream>>>(node_emb, gctx, head, X);
    k_ctx_gemm<<<dim3(16, 8), 128, 0, stream>>>(X, Wc, cap, dem, rem, SC);
    k_mha     <<<Bn, 512, 0, stream>>>(SC, gkey, gval, feas, m1, m2);
    k_gemm256 <<<dim3(16, 4, 2), 128, 0, stream>>>(m1, Wo, glimpse, m2, Wsa, henc);
    k_logits  <<<Bn, 512, 0, stream>>>(glimpse, lkey, feas, logp);
}